// GNNembedding_60112362275091
// MI455X (gfx1250) — compile-verified
//
#include <hip/hip_runtime.h>
#include <hip/hip_bf16.h>
#include <stdint.h>

// GraphSAGE single layer (PyG SAGEConv, mean aggregation) for MI455X / gfx1250.
//   Phase 0: zero workspace (summed[N,64], cnt[N])
//   Phase 1: scatter-add neighbor features + degree counts (hw fp32 atomics, L2-resident)
//   Phase 2: fused mean + dual GEMM via V_WMMA_F32_16X16X4_F32 (exact fp32 tensor path)

typedef __attribute__((ext_vector_type(2))) float v2f;
typedef __attribute__((ext_vector_type(8))) float v8f;

#define NODES  50000
#define INDIM  64
#define OUTDIM 128
#define EDGES  800000

__global__ void sage_zero_ws(float* __restrict__ ws, int n) {
    int i = blockIdx.x * blockDim.x + threadIdx.x;
    if (i < n) ws[i] = 0.0f;
}

// One 32-lane group per edge; each lane handles 2 of the 64 feature columns.
__global__ void sage_scatter(const float* __restrict__ x,
                             const long long* __restrict__ src,
                             const long long* __restrict__ dst,
                             float* __restrict__ summed,
                             float* __restrict__ cnt) {
    int tid  = blockIdx.x * blockDim.x + threadIdx.x;
    int e    = tid >> 5;
    int lane = tid & 31;
    if (e >= EDGES) return;
    int s = (int)src[e];
    int d = (int)dst[e];
    float2 v = ((const float2*)(x + (size_t)s * INDIM))[lane];
    float* drow = summed + (size_t)d * INDIM + lane * 2;
    unsafeAtomicAdd(drow + 0, v.x);   // global_atomic_add_f32 (no return)
    unsafeAtomicAdd(drow + 1, v.y);
    if (lane == 0) unsafeAtomicAdd(&cnt[d], 1.0f);
}

// Block = 256 threads = 8 waves. blockIdx.x selects a 16-row tile of nodes;
// wave id selects a 16-column tile of the 128 output columns.
// out[16x16] = (summed/clamp(cnt,1)) @ W_l^T  +  x @ W_r^T  +  b_l
__global__ void __launch_bounds__(256)
sage_gemm_wmma(const float* __restrict__ x,
               const float* __restrict__ summed,
               const float* __restrict__ cnt,
               const float* __restrict__ Wl,
               const float* __restrict__ bl,
               const float* __restrict__ Wr,
               float* __restrict__ out) {
    const int wave = threadIdx.x >> 5;       // 0..7  -> N tile
    const int lane = threadIdx.x & 31;
    const int half = lane >> 4;              // 0: K{0,1} / M 0..7 ; 1: K{2,3} / M 8..15
    const int l15  = lane & 15;
    const int m0   = blockIdx.x << 4;        // node row tile (50000 % 16 == 0)
    const int n0   = wave << 4;              // output col tile
    const int row  = m0 + l15;               // A-matrix row for this lane
    const int col  = n0 + l15;               // B/C/D column for this lane
    const int kk   = half << 1;              // K sub-offset within each K4 step

    const float invc = 1.0f / fmaxf(cnt[row], 1.0f);

    // Seed accumulator with bias (constant per output column).
    const float bias = bl[col];
    v8f c;
#pragma unroll
    for (int i = 0; i < 8; ++i) c[i] = bias;

    const float* arow_m = summed + (size_t)row * INDIM;  // mean path A rows
    const float* arow_x = x      + (size_t)row * INDIM;  // root path A rows
    const float* brow_l = Wl     + (size_t)col * INDIM;  // W_l^T[k][col] = Wl[col*64+k]
    const float* brow_r = Wr     + (size_t)col * INDIM;

    // GEMM 1: mean @ W_l^T   (16 x K4 steps, fp32 WMMA, exact arithmetic)
#pragma unroll
    for (int k0 = 0; k0 < INDIM; k0 += 4) {
        float2 av = *(const float2*)(arow_m + k0 + kk);
        float2 bv = *(const float2*)(brow_l + k0 + kk);
        v2f a, b;
        a.x = av.x * invc; a.y = av.y * invc;   // fold mean division into A load
        b.x = bv.x;        b.y = bv.y;
        c = __builtin_amdgcn_wmma_f32_16x16x4_f32(false, a, false, b,
                                                  (short)0, c, false, false);
    }
    // GEMM 2: x @ W_r^T
#pragma unroll
    for (int k0 = 0; k0 < INDIM; k0 += 4) {
        float2 av = *(const float2*)(arow_x + k0 + kk);
        float2 bv = *(const float2*)(brow_r + k0 + kk);
        v2f a, b;
        a.x = av.x; a.y = av.y;
        b.x = bv.x; b.y = bv.y;
        c = __builtin_amdgcn_wmma_f32_16x16x4_f32(false, a, false, b,
                                                  (short)0, c, false, false);
    }

    // D layout: VGPR v -> M = v + 8*half (lanes 16..31 hold M=8..15), N = col.
#pragma unroll
    for (int v = 0; v < 8; ++v) {
        out[(size_t)(m0 + (half << 3) + v) * OUTDIM + col] = c[v];
    }
}

extern "C" void kernel_launch(void* const* d_in, const int* in_sizes, int n_in,
                              void* d_out, int out_size, void* d_ws, size_t ws_size,
                              hipStream_t stream) {
    const float*     x  = (const float*)d_in[0];
    const long long* ei = (const long long*)d_in[1];   // [2, EDGES] int64
    const float*     Wl = (const float*)d_in[2];
    const float*     bl = (const float*)d_in[3];
    const float*     Wr = (const float*)d_in[4];
    float* out = (float*)d_out;

    float* summed = (float*)d_ws;                       // [NODES, INDIM]
    float* cnt    = summed + (size_t)NODES * INDIM;     // [NODES]

    const int nzero = NODES * INDIM + NODES;
    sage_zero_ws<<<(nzero + 255) / 256, 256, 0, stream>>>(summed, nzero);

    const long long nthreads = (long long)EDGES * 32;
    sage_scatter<<<(int)((nthreads + 255) / 256), 256, 0, stream>>>(
        x, ei, ei + EDGES, summed, cnt);

    sage_gemm_wmma<<<NODES / 16, 256, 0, stream>>>(x, summed, cnt, Wl, bl, Wr, out);
}